// Edge_Encoder_Residual_20813411516978
// MI455X (gfx1250) — compile-verified
//
#include <hip/hip_runtime.h>
#include <hip/hip_bf16.h>
#include <math.h>

// ---------------------------------------------------------------------------
// Problem constants (match the reference)
// ---------------------------------------------------------------------------
#define NN      100000      // nodes
#define EE      1600000     // edges (before self loops)
#define F_IN    16
#define HC      64          // heads*channels
#define HEADS   2
#define CHAN    32
#define TD      16
#define CD      16
#define NC      2

typedef __attribute__((ext_vector_type(2))) float v2f;
typedef __attribute__((ext_vector_type(8))) float v8f;

// ---------------------------------------------------------------------------
// Helpers
// ---------------------------------------------------------------------------
__device__ __forceinline__ void atomicMaxF(float* addr, float v) {
    // ordered-int trick: works for mixed-sign floats, init must be -INF
    if (v >= 0.0f) atomicMax((int*)addr, __float_as_int(v));
    else           atomicMin((unsigned int*)addr, (unsigned int)__float_as_int(v));
}

// ---------------------------------------------------------------------------
// Generic fp32 WMMA GEMM:  D[16x16 tile] = A[16xK] * B[Kx16] (+bias)(+relu)
// A row-major [M x lda], B row-major [K x ldb], D row-major [M x ldo]
// One wave (32 threads) per 16x16 output tile.
// VGPR layouts per CDNA5 ISA 7.12.2:
//   A/B component r: K = r + 2*(lane>=16);  A row = lane&15, B col = lane&15
//   C/D component r: row = r + 8*(lane>=16), col = lane&15
// ---------------------------------------------------------------------------
__global__ void k_gemm_wmma(const float* __restrict__ A, int lda,
                            const float* __restrict__ B, int ldb,
                            const float* __restrict__ bias,
                            float* __restrict__ D, int ldo,
                            int K, int do_relu) {
    const int lane = threadIdx.x & 31;
    const int m    = lane & 15;
    const int grp  = lane >> 4;
    const long rm  = (long)blockIdx.x * 16;
    const long cn  = (long)blockIdx.y * 16;

    v8f c = {};
    for (int k0 = 0; k0 < K; k0 += 4) {
        const int ka = k0 + 2 * grp;
        v2f a, b;
        a.x = A[(rm + m) * lda + ka];
        a.y = A[(rm + m) * lda + ka + 1];
        b.x = B[(long)ka       * ldb + cn + m];
        b.y = B[(long)(ka + 1) * ldb + cn + m];
        c = __builtin_amdgcn_wmma_f32_16x16x4_f32(false, a, false, b,
                                                  (short)0, c, false, false);
    }
    const long col = cn + m;
    const float bv = bias ? bias[col] : 0.0f;
#pragma unroll
    for (int r = 0; r < 8; ++r) {
        const long row = rm + r + 8 * grp;
        float v = c[r] + bv;
        if (do_relu) v = fmaxf(v, 0.0f);
        D[row * ldo + col] = v;
    }
}

// ---------------------------------------------------------------------------
// Decoder GEMM: A is the virtual concat [h(64) | t_emb(16) | cond(16)] (K=96),
// B = Wfd [96x32], out fd [N x 32] (+bfd). One wave per 16x16 tile.
// ---------------------------------------------------------------------------
__device__ __forceinline__ float dec_a(const float* __restrict__ h,
                                       const float* __restrict__ temb,
                                       const float* __restrict__ cond,
                                       long row, int k) {
    if (k < HC)      return h[row * HC + k];
    if (k < HC + TD) return temb[k - HC];
    return cond[row * CD + (k - HC - TD)];
}

__global__ void k_dec_gemm(const float* __restrict__ Hf,
                           const float* __restrict__ temb,
                           const float* __restrict__ cond,
                           const float* __restrict__ Wfd,
                           const float* __restrict__ bfd,
                           float* __restrict__ fd) {
    const int lane = threadIdx.x & 31;
    const int m    = lane & 15;
    const int grp  = lane >> 4;
    const long rm  = (long)blockIdx.x * 16;
    const long cn  = (long)blockIdx.y * 16;

    v8f c = {};
    for (int k0 = 0; k0 < HC + TD + CD; k0 += 4) {
        const int ka = k0 + 2 * grp;
        v2f a, b;
        a.x = dec_a(Hf, temb, cond, rm + m, ka);
        a.y = dec_a(Hf, temb, cond, rm + m, ka + 1);
        b.x = Wfd[(long)ka       * CHAN + cn + m];
        b.y = Wfd[(long)(ka + 1) * CHAN + cn + m];
        c = __builtin_amdgcn_wmma_f32_16x16x4_f32(false, a, false, b,
                                                  (short)0, c, false, false);
    }
    const long col = cn + m;
    const float bv = bfd[col];
#pragma unroll
    for (int r = 0; r < 8; ++r) {
        const long row = rm + r + 8 * grp;
        fd[row * CHAN + col] = c[r] + bv;
    }
}

// ---------------------------------------------------------------------------
// Per-call init of accumulators (harness does not re-zero ws between replays)
// ---------------------------------------------------------------------------
__global__ void k_init(float* __restrict__ acc, float* __restrict__ nmax,
                       float* __restrict__ nsum, int n64, int n2) {
    const int i = blockIdx.x * blockDim.x + threadIdx.x;
    if (i < n64) acc[i] = 0.0f;
    if (i < n2) { nmax[i] = -INFINITY; nsum[i] = 0.0f; }
}

// ---------------------------------------------------------------------------
// Edge pass 1: logits + segment max (self loops synthesized for e >= E)
// ---------------------------------------------------------------------------
__global__ void k_edge_logit(const float* __restrict__ xl,
                             const float* __restrict__ xr,
                             const float* __restrict__ att,   // [2,32]
                             const int* __restrict__ src,
                             const int* __restrict__ dst,
                             float* __restrict__ logit,
                             float* __restrict__ nmax) {
    const int e = blockIdx.x * blockDim.x + threadIdx.x;
    if (e >= EE + NN) return;
    const int s = (e < EE) ? src[e] : (e - EE);
    const int d = (e < EE) ? dst[e] : (e - EE);
    const float4* pl = (const float4*)(xl + (long)s * HC);
    const float4* pr = (const float4*)(xr + (long)d * HC);
    const float4* pa = (const float4*)att;
#pragma unroll
    for (int h = 0; h < HEADS; ++h) {
        float acc = 0.0f;
#pragma unroll
        for (int q = 0; q < 8; ++q) {
            const float4 a = pl[h * 8 + q];
            const float4 b = pr[h * 8 + q];
            const float4 w = pa[h * 8 + q];
            float v;
            v = a.x + b.x; v = v > 0.f ? v : 0.2f * v; acc += v * w.x;
            v = a.y + b.y; v = v > 0.f ? v : 0.2f * v; acc += v * w.y;
            v = a.z + b.z; v = v > 0.f ? v : 0.2f * v; acc += v * w.z;
            v = a.w + b.w; v = v > 0.f ? v : 0.2f * v; acc += v * w.w;
        }
        logit[(long)e * HEADS + h] = acc;
        atomicMaxF(&nmax[(long)d * HEADS + h], acc);
    }
}

// ---------------------------------------------------------------------------
// Edge pass 2: exp(logit - max) in place + segment sum
// ---------------------------------------------------------------------------
__global__ void k_edge_expsum(float* __restrict__ logit,
                              const int* __restrict__ dst,
                              const float* __restrict__ nmax,
                              float* __restrict__ nsum) {
    const int e = blockIdx.x * blockDim.x + threadIdx.x;
    if (e >= EE + NN) return;
    const int d = (e < EE) ? dst[e] : (e - EE);
#pragma unroll
    for (int h = 0; h < HEADS; ++h) {
        const float ex = __expf(logit[(long)e * HEADS + h] -
                                nmax[(long)d * HEADS + h]);
        logit[(long)e * HEADS + h] = ex;
        atomicAdd(&nsum[(long)d * HEADS + h], ex);
    }
}

// ---------------------------------------------------------------------------
// Edge pass 3: scatter messages  acc[dst] += xl[src] * alpha
// ---------------------------------------------------------------------------
__global__ void k_edge_msg(const float* __restrict__ xl,
                           const float* __restrict__ logit,
                           const float* __restrict__ nsum,
                           const int* __restrict__ src,
                           const int* __restrict__ dst,
                           float* __restrict__ acc) {
    const int e = blockIdx.x * blockDim.x + threadIdx.x;
    if (e >= EE + NN) return;
    const int s = (e < EE) ? src[e] : (e - EE);
    const int d = (e < EE) ? dst[e] : (e - EE);
#pragma unroll
    for (int h = 0; h < HEADS; ++h) {
        const float alpha = logit[(long)e * HEADS + h] /
                            (nsum[(long)d * HEADS + h] + 1e-16f);
        const float* p = xl + (long)s * HC + h * CHAN;
        float* o       = acc + (long)d * HC + h * CHAN;
#pragma unroll
        for (int c = 0; c < CHAN; ++c) atomicAdd(&o[c], p[c] * alpha);
    }
}

// ---------------------------------------------------------------------------
// Finalize layer: h = relu(acc + bias) * THETA(=1) + res(already relu'd)
// ---------------------------------------------------------------------------
__global__ void k_finalize(const float* __restrict__ acc,
                           const float* __restrict__ bias,   // nullable, [64]
                           const float* __restrict__ res,
                           float* __restrict__ hout, int total) {
    const int i = blockIdx.x * blockDim.x + threadIdx.x;
    if (i >= total) return;
    float v = acc[i] + (bias ? bias[i & (HC - 1)] : 0.0f);
    hout[i] = fmaxf(v, 0.0f) + res[i];
}

// ---------------------------------------------------------------------------
// Time embedding + 2-layer SiLU MLP (tiny, one wave)
// ---------------------------------------------------------------------------
__global__ void k_time(const int* __restrict__ t,
                       const float* __restrict__ Wt0, const float* __restrict__ bt0,
                       const float* __restrict__ Wt1, const float* __restrict__ bt1,
                       float* __restrict__ temb_out) {
    __shared__ float e0[TD], e1[TD];
    const int i = threadIdx.x;
    const float tf = (float)t[0];          // * (1000/MAX_TIME) == 1
    if (i < TD / 2) {
        const float freq = __expf((float)i * -(__logf(10000.0f) / (TD / 2 - 1)));
        const float ang = tf * freq;
        e0[i] = __sinf(ang);
        e0[i + TD / 2] = __cosf(ang);
    }
    __syncthreads();
    if (i < TD) {
        float a = bt0[i];
        for (int k = 0; k < TD; ++k) a += e0[k] * Wt0[k * TD + i];
        e1[i] = a / (1.0f + __expf(-a));   // silu
    }
    __syncthreads();
    if (i < TD) {
        float a = bt1[i];
        for (int k = 0; k < TD; ++k) a += e1[k] * Wt1[k * TD + i];
        temb_out[i] = a / (1.0f + __expf(-a));
    }
}

// ---------------------------------------------------------------------------
// Final classifier: logits[n] = fd[n,32] @ Wcls[32x2] + bcls
// ---------------------------------------------------------------------------
__global__ void k_cls(const float* __restrict__ fd,
                      const float* __restrict__ Wcls,
                      const float* __restrict__ bcls,
                      float* __restrict__ out) {
    const int n = blockIdx.x * blockDim.x + threadIdx.x;
    if (n >= NN) return;
    const float* f = fd + (long)n * CHAN;
    float o0 = bcls[0], o1 = bcls[1];
#pragma unroll
    for (int c = 0; c < CHAN; ++c) {
        const float v = f[c];
        o0 += v * Wcls[c * NC + 0];
        o1 += v * Wcls[c * NC + 1];
    }
    out[(long)n * NC + 0] = o0;
    out[(long)n * NC + 1] = o1;
}

// ---------------------------------------------------------------------------
// Host-side orchestration
// ---------------------------------------------------------------------------
extern "C" void kernel_launch(void* const* d_in, const int* in_sizes, int n_in,
                              void* d_out, int out_size, void* d_ws, size_t ws_size,
                              hipStream_t stream) {
    // Inputs in setup_inputs() order
    const float* x     = (const float*)d_in[0];
    const int*   ei    = (const int*)d_in[1];     // [2,E] flat
    const int*   t     = (const int*)d_in[2];
    const float* cond  = (const float*)d_in[3];
    const float* Wl0   = (const float*)d_in[4];
    const float* bl0   = (const float*)d_in[5];
    const float* Wr0   = (const float*)d_in[6];
    const float* br0   = (const float*)d_in[7];
    const float* att0  = (const float*)d_in[8];
    const float* bias0 = (const float*)d_in[9];
    const float* Wres0 = (const float*)d_in[10];
    const float* bres0 = (const float*)d_in[11];
    const float* Wl1   = (const float*)d_in[12];
    const float* Wr1   = (const float*)d_in[13];
    const float* att1  = (const float*)d_in[14];
    const float* Wres1 = (const float*)d_in[15];
    const float* bres1 = (const float*)d_in[16];
    const float* Wt0   = (const float*)d_in[17];
    const float* bt0   = (const float*)d_in[18];
    const float* Wt1   = (const float*)d_in[19];
    const float* bt1   = (const float*)d_in[20];
    const float* Wfd   = (const float*)d_in[21];
    const float* bfd   = (const float*)d_in[22];
    const float* Wcls  = (const float*)d_in[23];
    const float* bcls  = (const float*)d_in[24];

    const int* src = ei;
    const int* dst = ei + EE;

    // Workspace layout (floats)
    float* ws = (float*)d_ws;
    const long NB = (long)NN * HC;                // 6.4M floats per node buffer
    float* bufXL  = ws;                           // xl
    float* bufXR  = ws + 1 * NB;                  // xr
    float* bufRES = ws + 2 * NB;                  // residual
    float* bufACC = ws + 3 * NB;                  // scatter accumulator
    float* bufH   = ws + 4 * NB;                  // layer output h
    float* logit  = ws + 5 * NB;                  // (E+N)*2
    float* nmax   = logit + (long)(EE + NN) * HEADS;
    float* nsum   = nmax + (long)NN * HEADS;
    float* temb   = nsum + (long)NN * HEADS;      // 16 floats
    float* bufFD  = bufXL;                        // decoder fd reuses xl (free then)

    const dim3 b32(32);
    const dim3 gGemm64(NN / 16, HC / 16);         // 6250 x 4 tiles
    const int EB = 256;
    const int edgeBlocks = (EE + NN + EB - 1) / EB;
    const int initBlocks = ((int)NB + EB - 1) / EB;

    // --- time MLP (independent) ---
    k_time<<<1, 32, 0, stream>>>(t, Wt0, bt0, Wt1, bt1, temb);

    // ======================= layer 0 =======================
    k_gemm_wmma<<<gGemm64, b32, 0, stream>>>(x, F_IN, Wl0,   HC, bl0,   bufXL,  HC, F_IN, 0);
    k_gemm_wmma<<<gGemm64, b32, 0, stream>>>(x, F_IN, Wr0,   HC, br0,   bufXR,  HC, F_IN, 0);
    k_gemm_wmma<<<gGemm64, b32, 0, stream>>>(x, F_IN, Wres0, HC, bres0, bufRES, HC, F_IN, 1);
    k_init<<<initBlocks, EB, 0, stream>>>(bufACC, nmax, nsum, (int)NB, NN * HEADS);
    k_edge_logit <<<edgeBlocks, EB, 0, stream>>>(bufXL, bufXR, att0, src, dst, logit, nmax);
    k_edge_expsum<<<edgeBlocks, EB, 0, stream>>>(logit, dst, nmax, nsum);
    k_edge_msg   <<<edgeBlocks, EB, 0, stream>>>(bufXL, logit, nsum, src, dst, bufACC);
    k_finalize<<<initBlocks, EB, 0, stream>>>(bufACC, bias0, bufRES, bufH, (int)NB);

    // ======================= layer 1 =======================
    k_gemm_wmma<<<gGemm64, b32, 0, stream>>>(bufH, HC, Wl1,   HC, nullptr, bufXL,  HC, HC, 0);
    k_gemm_wmma<<<gGemm64, b32, 0, stream>>>(bufH, HC, Wr1,   HC, nullptr, bufXR,  HC, HC, 0);
    k_gemm_wmma<<<gGemm64, b32, 0, stream>>>(bufH, HC, Wres1, HC, bres1,   bufRES, HC, HC, 1);
    k_init<<<initBlocks, EB, 0, stream>>>(bufACC, nmax, nsum, (int)NB, NN * HEADS);
    k_edge_logit <<<edgeBlocks, EB, 0, stream>>>(bufXL, bufXR, att1, src, dst, logit, nmax);
    k_edge_expsum<<<edgeBlocks, EB, 0, stream>>>(logit, dst, nmax, nsum);
    k_edge_msg   <<<edgeBlocks, EB, 0, stream>>>(bufXL, logit, nsum, src, dst, bufACC);
    k_finalize<<<initBlocks, EB, 0, stream>>>(bufACC, nullptr, bufRES, bufH, (int)NB);

    // ======================= decoder =======================
    const dim3 gDec(NN / 16, CHAN / 16);          // 6250 x 2
    k_dec_gemm<<<gDec, b32, 0, stream>>>(bufH, temb, cond, Wfd, bfd, bufFD);
    k_cls<<<(NN + EB - 1) / EB, EB, 0, stream>>>(bufFD, Wcls, bcls, (float*)d_out);
}